// TemporalFlowCell_154618823405
// MI455X (gfx1250) — compile-verified
//
#include <hip/hip_runtime.h>
#include <math.h>

typedef float v2f __attribute__((ext_vector_type(2)));
typedef float v8f __attribute__((ext_vector_type(8)));

#define T_LEN 4096
#define B_SZ  8
#define D_SZ  512
#define K_SZ  64

static __device__ __forceinline__ v8f wmma_f32(v2f a, v2f b, v8f c) {
    return __builtin_amdgcn_wmma_f32_16x16x4_f32(false, a, false, b, (short)0, c,
                                                 false, false);
}

// ---------------------------------------------------------------------------
// Kernel 1: beta[t][b][k] = sum_d x[b][t][d] * W_in[k][d]
// rows r = t*8+b (32768 rows). One wave per 16-row M-tile, full K=64 output.
// ---------------------------------------------------------------------------
__global__ __launch_bounds__(256) void beta_gemm_kernel(const float* __restrict__ x,
                                                        const float* __restrict__ Win,
                                                        float* __restrict__ betaw) {
    const int tid  = threadIdx.x;
    const int wid  = tid >> 5;
    const int lane = tid & 31;
    const int ln   = lane & 15;
    const int half = lane >> 4;

    const int wg = blockIdx.x * 8 + wid;   // M-tile index 0..2047
    const int r  = wg * 16 + ln;           // this lane's A row
    const int b_ = r & 7;
    const int t_ = r >> 3;
    const float* xrow = x + ((size_t)b_ * T_LEN + t_) * D_SZ;

    v8f acc[4];
#pragma unroll
    for (int nt = 0; nt < 4; nt++) acc[nt] = (v8f){0,0,0,0,0,0,0,0};

#pragma unroll 8
    for (int kk = 0; kk < D_SZ / 4; kk++) {
        v2f a = *(const v2f*)(xrow + 4 * kk + 2 * half);
#pragma unroll
        for (int nt = 0; nt < 4; nt++) {
            int n   = nt * 16 + ln;
            v2f bfr = *(const v2f*)(Win + (size_t)n * D_SZ + 4 * kk + 2 * half);
            acc[nt] = wmma_f32(a, bfr, acc[nt]);
        }
    }

#pragma unroll
    for (int nt = 0; nt < 4; nt++) {
#pragma unroll
        for (int i = 0; i < 8; i++) {
            int row = wg * 16 + i + 8 * half;
            betaw[(size_t)row * K_SZ + nt * 16 + ln] = acc[nt][i];
        }
    }
}

// ---------------------------------------------------------------------------
// Kernel 2: serial scan.  State stacked [16 x 64] (rows 0-7 = sr, 8-15 = si).
// One workgroup, 4 waves; wave j owns output columns 16j..16j+15.
// Per step: A = rotate(prev state) + beta (pure FMA from prefetched regs),
// two 8-deep WMMA chains (K=64), publish tile to double-buffered LDS, one
// barrier.  Next step's beta fragments prefetched during the WMMA drain.
// ---------------------------------------------------------------------------
__global__ __launch_bounds__(128) void scan_kernel(const float* __restrict__ betaw,
                                                   const float* __restrict__ alpha,
                                                   const float* __restrict__ omega,
                                                   const float* __restrict__ resonance,
                                                   float* __restrict__ srseq,
                                                   float* __restrict__ stateTail) {
    __shared__ float S[2][16 * 68];   // padded rows: stride 68 floats

    const int tid  = threadIdx.x;
    const int wid  = tid >> 5;
    const int lane = tid & 31;
    const int ln   = lane & 15;
    const int half = lane >> 4;
    const int n0   = wid * 16;        // this wave's output column base
    const int coff = 2 * half;        // K offset within each 4-wide k-group
    const int bsel = ln & 7;          // state batch row this lane reads
    const bool isReal = (ln < 8);
    const float bEn   = isReal ? 1.f : 0.f;

    // zero initial state buffer
    for (int i = tid; i < 16 * 68; i += 128) S[0][i] = 0.f;

    // hoisted per-lane loop invariants: rotation coeffs + resonance B-fragments
    v2f p[16], q[16], Bf[16];
#pragma unroll
    for (int kk = 0; kk < 16; kk++) {
        int k0 = 4 * kk + coff;
        float m0 = 1.f / (1.f + expf(-alpha[k0]));
        float m1 = 1.f / (1.f + expf(-alpha[k0 + 1]));
        float c0 = cosf(omega[k0]),     s0 = sinf(omega[k0]);
        float c1 = cosf(omega[k0 + 1]), s1 = sinf(omega[k0 + 1]);
        float mc0 = m0 * c0, ms0 = m0 * s0;
        float mc1 = m1 * c1, ms1 = m1 * s1;
        p[kk].x = isReal ? mc0 : ms0;   q[kk].x = isReal ? -ms0 : mc0;
        p[kk].y = isReal ? mc1 : ms1;   q[kk].y = isReal ? -ms1 : mc1;
        Bf[kk].x = resonance[(size_t)k0 * K_SZ + n0 + ln];
        Bf[kk].y = resonance[(size_t)(k0 + 1) * K_SZ + n0 + ln];
    }

    // prefetch beta for t=0, pre-masked by real/imag row type
    v2f bv[16];
    {
        const float* bp = betaw + (size_t)bsel * K_SZ;
#pragma unroll
        for (int kk = 0; kk < 16; kk++) {
            v2f tv   = *(const v2f*)(bp + 4 * kk + coff);
            bv[kk].x = tv.x * bEn;
            bv[kk].y = tv.y * bEn;
        }
    }
    __syncthreads();

    v8f accOut = (v8f){0,0,0,0,0,0,0,0};

    for (int t = 0; t < T_LEN; t++) {
        const int buf  = t & 1;
        const int nbuf = buf ^ 1;

        v8f accA = (v8f){0,0,0,0,0,0,0,0};
        v8f accB = (v8f){0,0,0,0,0,0,0,0};

#pragma unroll
        for (int kk = 0; kk < 8; kk++) {
            int c  = 4 * kk + coff;
            v2f sr = *(const v2f*)&S[buf][bsel * 68 + c];
            v2f si = *(const v2f*)&S[buf][(bsel + 8) * 68 + c];
            v2f a;
            a.x = fmaf(p[kk].x, sr.x, fmaf(q[kk].x, si.x, bv[kk].x));
            a.y = fmaf(p[kk].y, sr.y, fmaf(q[kk].y, si.y, bv[kk].y));
            accA = wmma_f32(a, Bf[kk], accA);
        }
#pragma unroll
        for (int kk = 8; kk < 16; kk++) {
            int c  = 4 * kk + coff;
            v2f sr = *(const v2f*)&S[buf][bsel * 68 + c];
            v2f si = *(const v2f*)&S[buf][(bsel + 8) * 68 + c];
            v2f a;
            a.x = fmaf(p[kk].x, sr.x, fmaf(q[kk].x, si.x, bv[kk].x));
            a.y = fmaf(p[kk].y, sr.y, fmaf(q[kk].y, si.y, bv[kk].y));
            accB = wmma_f32(a, Bf[kk], accB);
        }

        // prefetch next step's beta while the WMMA chains drain
        if (t + 1 < T_LEN) {
            const float* bp = betaw + (size_t)(t + 1) * (B_SZ * K_SZ)
                            + (size_t)bsel * K_SZ;
#pragma unroll
            for (int kk = 0; kk < 16; kk++) {
                v2f tv   = *(const v2f*)(bp + 4 * kk + coff);
                bv[kk].x = tv.x * bEn;
                bv[kk].y = tv.y * bEn;
            }
        }

        accOut = accA + accB;

        // publish new state tile (columns n0..n0+15, rows via C layout)
#pragma unroll
        for (int i = 0; i < 8; i++)
            S[nbuf][(i + 8 * half) * 68 + n0 + ln] = accOut[i];

        // stream sr rows (lanes 0-15 hold all real rows) to sr_seq[t][b][k]
        if (half == 0) {
            float* sp = srseq + (size_t)t * (B_SZ * K_SZ) + n0 + ln;
#pragma unroll
            for (int i = 0; i < 8; i++) sp[i * K_SZ] = accOut[i];
        }
        __syncthreads();
    }

    // final states: real then imag, each [8][64]
#pragma unroll
    for (int i = 0; i < 8; i++) {
        if (half == 0) stateTail[i * K_SZ + n0 + ln]       = accOut[i];
        else           stateTail[512 + i * K_SZ + n0 + ln] = accOut[i];
    }
}

// ---------------------------------------------------------------------------
// Kernel 3: y = sr_seq . W_out^T  fused with LayerNorm over D=512.
// One block per 16-row M-tile; 8 waves x 64 cols cover the full row.
// ---------------------------------------------------------------------------
__global__ __launch_bounds__(256) void out_ln_kernel(const float* __restrict__ srseq,
                                                     const float* __restrict__ Wout,
                                                     const float* __restrict__ lng,
                                                     const float* __restrict__ lnb,
                                                     float* __restrict__ out) {
    __shared__ float sums[16];
    __shared__ float sumsq[16];

    const int tid  = threadIdx.x;
    const int wid  = tid >> 5;
    const int lane = tid & 31;
    const int ln   = lane & 15;
    const int half = lane >> 4;
    const int mtile = blockIdx.x;

    if (tid < 16) { sums[tid] = 0.f; sumsq[tid] = 0.f; }
    __syncthreads();

    const int r = mtile * 16 + ln;
    const float* arow = srseq + (size_t)r * K_SZ;

    v8f acc[4];
#pragma unroll
    for (int nt = 0; nt < 4; nt++) acc[nt] = (v8f){0,0,0,0,0,0,0,0};

#pragma unroll
    for (int kk = 0; kk < K_SZ / 4; kk++) {
        v2f a = *(const v2f*)(arow + 4 * kk + 2 * half);
#pragma unroll
        for (int nt = 0; nt < 4; nt++) {
            int d  = wid * 64 + nt * 16 + ln;
            v2f bf = *(const v2f*)(Wout + (size_t)d * K_SZ + 4 * kk + 2 * half);
            acc[nt] = wmma_f32(a, bf, acc[nt]);
        }
    }

    // per-row partial sums / sums of squares over this wave's 64 columns
#pragma unroll
    for (int i = 0; i < 8; i++) {
        float sv = 0.f, qv = 0.f;
#pragma unroll
        for (int nt = 0; nt < 4; nt++) {
            float v = acc[nt][i];
            sv += v;
            qv += v * v;
        }
#pragma unroll
        for (int m = 1; m < 16; m <<= 1) {
            sv += __shfl_xor(sv, m, 32);
            qv += __shfl_xor(qv, m, 32);
        }
        if (ln == 0) {
            atomicAdd(&sums[i + 8 * half], sv);
            atomicAdd(&sumsq[i + 8 * half], qv);
        }
    }
    __syncthreads();

    const float inv_d = 1.f / (float)D_SZ;
#pragma unroll
    for (int nt = 0; nt < 4; nt++) {
        int col  = wid * 64 + nt * 16 + ln;
        float gg = lng[col];
        float bb = lnb[col];
#pragma unroll
        for (int i = 0; i < 8; i++) {
            int row   = i + 8 * half;
            float mu  = sums[row] * inv_d;
            float var = sumsq[row] * inv_d - mu * mu;
            float rs  = rsqrtf(var + 1e-5f);
            int grow  = mtile * 16 + row;       // row index in (t,b) order
            int b_ = grow & 7, t_ = grow >> 3;
            out[((size_t)b_ * T_LEN + t_) * D_SZ + col] =
                (acc[nt][i] - mu) * rs * gg + bb;
        }
    }
}

// ---------------------------------------------------------------------------
extern "C" void kernel_launch(void* const* d_in, const int* in_sizes, int n_in,
                              void* d_out, int out_size, void* d_ws, size_t ws_size,
                              hipStream_t stream) {
    const float* x      = (const float*)d_in[0];
    const float* alpha  = (const float*)d_in[1];
    const float* omega  = (const float*)d_in[2];
    const float* W_in   = (const float*)d_in[3];
    const float* W_out  = (const float*)d_in[4];
    const float* reso   = (const float*)d_in[5];
    const float* lng    = (const float*)d_in[6];
    const float* lnb    = (const float*)d_in[7];

    float* out = (float*)d_out;
    float* stateTail = out + (size_t)B_SZ * T_LEN * D_SZ;   // [512 real][512 imag]

    float* betaw = (float*)d_ws;                              // [T][B][K] = 8 MB
    float* srseq = betaw + (size_t)T_LEN * B_SZ * K_SZ;       // [T][B][K] = 8 MB

    // 1) input projection GEMM: 2048 M-tiles, 8 waves/block
    beta_gemm_kernel<<<256, 256, 0, stream>>>(x, W_in, betaw);

    // 2) serial scan: one workgroup, 4 waves
    scan_kernel<<<1, 128, 0, stream>>>(betaw, alpha, omega, reso, srseq, stateTail);

    // 3) output projection + LayerNorm: one block per 16-row M-tile
    out_ln_kernel<<<2048, 256, 0, stream>>>(srseq, W_out, lng, lnb, out);
}